// AE_19052474925341
// MI455X (gfx1250) — compile-verified
//
#include <hip/hip_runtime.h>
#include <hip/hip_bf16.h>
#include <math.h>

// ---------------------------------------------------------------------------
// Spiral-conv mesh VAE decoder for MI455X (gfx1250, wave32, WMMA).
// All GEMMs use V_WMMA_F32_16X16X4_F32 (exact fp32 matrix pipe).
// ---------------------------------------------------------------------------

typedef float v2f __attribute__((ext_vector_type(2)));
typedef float v8f __attribute__((ext_vector_type(8)));

#define WMMA_F32_K4(a, b, c) \
    __builtin_amdgcn_wmma_f32_16x16x4_f32(false, (a), false, (b), (short)0, (c), false, false)

// dims
#define NB 8
#define LAT 64
#define V0 20000
#define V1 5000
#define V2 1250
#define V3 320
#define V4 80
#define SEQ 9

// ---------------------------------------------------------------------------
// z = mu + eps * exp(0.5*logvar); writes a ZERO-PADDED [16,64] z staging
// buffer (rows 8..15 = 0) so fc_kernel's A loads are EXEC-uniform.
// Also emits mu, logvar into d_out.
// ---------------------------------------------------------------------------
__global__ void z_kernel(const int* __restrict__ idx, const float* __restrict__ eps,
                         const float* __restrict__ mu_t, const float* __restrict__ lv_t,
                         float* __restrict__ z, float* __restrict__ out_mu,
                         float* __restrict__ out_lv) {
    int t = blockIdx.x * blockDim.x + threadIdx.x;
    if (t >= 16 * LAT) return;
    if (t < NB * LAT) {
        int b = t >> 6, l = t & 63;
        int row = idx[b];
        float m = mu_t[row * LAT + l];
        float lv = lv_t[row * LAT + l];
        z[t] = m + eps[t] * expf(0.5f * lv);
        out_mu[t] = m;
        out_lv[t] = lv;
    } else {
        z[t] = 0.f;  // padded rows 8..15
    }
}

// ---------------------------------------------------------------------------
// fc: out[8, 20480] = z[8,64] @ W[64,20480] + b  (M padded to 16, WMMA)
// one wave per 16-col N-tile, grid = 20480/16 = 1280. z is pre-padded.
// ---------------------------------------------------------------------------
__global__ __launch_bounds__(32) void fc_kernel(const float* __restrict__ z,
                                                const float* __restrict__ w,
                                                const float* __restrict__ bias,
                                                float* __restrict__ out) {
    const int lane = threadIdx.x & 31;
    const int n0 = blockIdx.x * 16;
    const int laneN = lane & 15;         // N for B / M-row for A
    const int khalf = (lane >> 4) << 1;  // 0 or 2
    v8f acc = {0.f, 0.f, 0.f, 0.f, 0.f, 0.f, 0.f, 0.f};
#pragma unroll 4
    for (int k = 0; k < LAT; k += 4) {
        v2f a, bf;
        a.x = z[laneN * LAT + k + khalf];
        a.y = z[laneN * LAT + k + khalf + 1];
        const float* wp = w + (size_t)(k + khalf) * 20480 + n0 + laneN;
        bf.x = wp[0];
        bf.y = wp[20480];
        acc = WMMA_F32_K4(a, bf, acc);
    }
    // lanes 0-15 hold rows M=0..7 (the valid ones); lanes 16-31 hold padded rows
    if (lane < 16) {
        float bn = bias[n0 + lane];
#pragma unroll
        for (int r = 0; r < 8; ++r)
            out[(size_t)r * 20480 + n0 + lane] = acc[r] + bn;
    }
}

// ---------------------------------------------------------------------------
// pool: out[b,v,c] = sum_{t<3} vals[3v+t] * x[b, cols[3v+t], c]
// (rows = repeat(arange(Vout),3) => pure gather, no atomics)
// ---------------------------------------------------------------------------
template <int C>
__global__ void pool_kernel(const float* __restrict__ x, const int* __restrict__ cols,
                            const float* __restrict__ vals, float* __restrict__ out,
                            int Vout, int Vin) {
    long long id = (long long)blockIdx.x * blockDim.x + threadIdx.x;
    long long total = (long long)NB * Vout * C;
    if (id >= total) return;
    int c = (int)(id & (C - 1));
    long long vb = id >> __builtin_ctz(C);
    int v = (int)(vb % Vout);
    int b = (int)(vb / Vout);
    const float* xb = x + (long long)b * Vin * C;
    int i = 3 * v;
    float s = vals[i]     * xb[(long long)cols[i]     * C + c]
            + vals[i + 1] * xb[(long long)cols[i + 1] * C + c]
            + vals[i + 2] * xb[(long long)cols[i + 2] * C + c];
    out[id] = s;
}

// ---------------------------------------------------------------------------
// spiral conv: out[B*V, COUT] = elu( gather(x, sp)[B*V, 9*CIN] @ W + b )
// block = (COUT/16) waves; block owns one 32-row M-tile (two 16-row WMMA
// tiles -> each B fragment feeds 2 WMMAs), wave owns one 16-col N-tile.
// A panel (32 rows x CIN per spiral tap) staged in LDS; B streamed from L2.
// ---------------------------------------------------------------------------
template <int CIN, int COUT, int VCUR, bool ELU>
__global__ __launch_bounds__((COUT / 16) * 32) void spiral_conv(
    const float* __restrict__ x, const int* __restrict__ sp,
    const float* __restrict__ w, const float* __restrict__ bias,
    float* __restrict__ out) {
    constexpr int NT = COUT / 16;
    constexpr int NTHR = NT * 32;
    constexpr int LDA = CIN + 4;       // stride%64==4 -> conflict-free column reads
    constexpr int MTOT = NB * VCUR;
    constexpr bool TAIL = (MTOT % 32) != 0;
    __shared__ float As[32 * LDA];
    __shared__ long long rowBase[32];

    const int tid = threadIdx.x;
    const int lane = tid & 31;
    const int wave = tid >> 5;
    const int n0 = wave * 16;
    const int laneN = lane & 15;         // N col for B, M row for A
    const int khalf = (lane >> 4) << 1;  // 0 or 2

    v8f acc0 = {0.f, 0.f, 0.f, 0.f, 0.f, 0.f, 0.f, 0.f};
    v8f acc1 = {0.f, 0.f, 0.f, 0.f, 0.f, 0.f, 0.f, 0.f};

    for (int j = 0; j < SEQ; ++j) {
        __syncthreads();  // previous iteration's LDS reads done
        if (tid < 32) {
            int g = blockIdx.x * 32 + tid;
            if (TAIL) g = g < MTOT ? g : MTOT - 1;  // clamp; EXEC stays uniform
            int b = g / VCUR;
            int v = g - b * VCUR;
            rowBase[tid] = ((long long)b * VCUR + sp[v * SEQ + j]) * CIN;
        }
        __syncthreads();
        // cooperative gather of the 32-row A panel (coalesced over c)
#pragma unroll
        for (int e = tid; e < 32 * CIN; e += NTHR) {
            int r = e / CIN;
            int c = e & (CIN - 1);
            As[r * LDA + c] = x[rowBase[r] + c];
        }
        __syncthreads();

        const float* wj = w + (size_t)(j * CIN) * COUT + n0 + laneN;
#pragma unroll 4
        for (int k = 0; k < CIN; k += 4) {
            v2f a0, a1, bf;
            a0.x = As[laneN * LDA + k + khalf];
            a0.y = As[laneN * LDA + k + khalf + 1];
            a1.x = As[(16 + laneN) * LDA + k + khalf];
            a1.y = As[(16 + laneN) * LDA + k + khalf + 1];
            const float* wp = wj + (size_t)(k + khalf) * COUT;
            bf.x = wp[0];
            bf.y = wp[COUT];
            acc0 = WMMA_F32_K4(a0, bf, acc0);  // independent chains alternate ->
            acc1 = WMMA_F32_K4(a1, bf, acc1);  // no back-to-back RAW on the XDL
        }
    }

    const float bn = bias[n0 + laneN];
    const int rbase = (lane >> 4) * 8;  // lanes 16-31 hold M=8..15 of each tile
    const long long grow0 = (long long)blockIdx.x * 32;
#pragma unroll
    for (int r = 0; r < 8; ++r) {
        long long g0 = grow0 + rbase + r;
        long long g1 = grow0 + 16 + rbase + r;
        float v0a = acc0[r] + bn;
        float v1a = acc1[r] + bn;
        if (ELU) {
            v0a = (v0a > 0.f) ? v0a : (expf(v0a) - 1.f);
            v1a = (v1a > 0.f) ? v1a : (expf(v1a) - 1.f);
        }
        if (!TAIL || g0 < MTOT) out[g0 * COUT + n0 + laneN] = v0a;
        if (!TAIL || g1 < MTOT) out[g1 * COUT + n0 + laneN] = v1a;
    }
}

// ---------------------------------------------------------------------------
// final conv: [B*V0, 9*32] @ W[288,3] + b  (tiny N -> scalar VALU tail)
// ---------------------------------------------------------------------------
__global__ void convf_kernel(const float* __restrict__ x, const int* __restrict__ sp0,
                             const float* __restrict__ w, const float* __restrict__ bias,
                             float* __restrict__ out) {
    int id = blockIdx.x * blockDim.x + threadIdx.x;
    if (id >= NB * V0) return;
    int b = id / V0, v = id - b * V0;
    float s0 = bias[0], s1 = bias[1], s2 = bias[2];
    const float* xb = x + (size_t)b * V0 * 32;
    for (int j = 0; j < SEQ; ++j) {
        const float* xr = xb + (size_t)sp0[v * SEQ + j] * 32;
        const float* wr = w + j * 32 * 3;
#pragma unroll
        for (int c = 0; c < 32; ++c) {
            float xv = xr[c];
            s0 = fmaf(xv, wr[c * 3 + 0], s0);
            s1 = fmaf(xv, wr[c * 3 + 1], s1);
            s2 = fmaf(xv, wr[c * 3 + 2], s2);
        }
    }
    size_t o = (size_t)id * 3;
    out[o] = s0; out[o + 1] = s1; out[o + 2] = s2;
}

// ---------------------------------------------------------------------------
// tiny MLP head with training-mode BatchNorm over B=8, leaky_relu(0.2)
// ---------------------------------------------------------------------------
__global__ void head_kernel(const float* __restrict__ z, int col,
                            const float* __restrict__ w1, const float* __restrict__ b1,
                            const float* __restrict__ g1, const float* __restrict__ be1,
                            const float* __restrict__ w2, const float* __restrict__ b2,
                            const float* __restrict__ g2, const float* __restrict__ be2,
                            const float* __restrict__ w3, const float* __restrict__ b3,
                            float* __restrict__ out, int do_sigmoid) {
    if (threadIdx.x != 0 || blockIdx.x != 0) return;
    float h[8][8], h2[8][8];
    for (int s = 0; s < 8; ++s)
        for (int j = 0; j < 8; ++j) h[s][j] = z[s * LAT + col] * w1[j] + b1[j];
    for (int j = 0; j < 8; ++j) {
        float mu = 0.f;
        for (int s = 0; s < 8; ++s) mu += h[s][j];
        mu *= 0.125f;
        float var = 0.f;
        for (int s = 0; s < 8; ++s) { float d = h[s][j] - mu; var += d * d; }
        var *= 0.125f;
        float inv = rsqrtf(var + 1e-5f);
        for (int s = 0; s < 8; ++s) {
            float hv = (h[s][j] - mu) * inv * g1[j] + be1[j];
            h[s][j] = hv > 0.f ? hv : 0.2f * hv;
        }
    }
    for (int s = 0; s < 8; ++s)
        for (int j = 0; j < 8; ++j) {
            float a = b2[j];
            for (int i = 0; i < 8; ++i) a += h[s][i] * w2[i * 8 + j];
            h2[s][j] = a;
        }
    for (int j = 0; j < 8; ++j) {
        float mu = 0.f;
        for (int s = 0; s < 8; ++s) mu += h2[s][j];
        mu *= 0.125f;
        float var = 0.f;
        for (int s = 0; s < 8; ++s) { float d = h2[s][j] - mu; var += d * d; }
        var *= 0.125f;
        float inv = rsqrtf(var + 1e-5f);
        for (int s = 0; s < 8; ++s) {
            float hv = (h2[s][j] - mu) * inv * g2[j] + be2[j];
            h2[s][j] = hv > 0.f ? hv : 0.2f * hv;
        }
    }
    for (int s = 0; s < 8; ++s) {
        float a = b3[0];
        for (int i = 0; i < 8; ++i) a += h2[s][i] * w3[i];
        if (do_sigmoid) a = 1.f / (1.f + expf(-a));
        out[s] = a;
    }
}

// ---------------------------------------------------------------------------
extern "C" void kernel_launch(void* const* d_in, const int* in_sizes, int n_in,
                              void* d_out, int out_size, void* d_ws, size_t ws_size,
                              hipStream_t stream) {
    const int*   idx      = (const int*)d_in[0];
    const float* eps      = (const float*)d_in[1];
    const float* mu_t     = (const float*)d_in[2];
    const float* lv_t     = (const float*)d_in[3];
    const float* fc_w     = (const float*)d_in[4];
    const float* fc_b     = (const float*)d_in[5];
    const int*   sp0      = (const int*)d_in[6];
    const int*   sp1      = (const int*)d_in[7];
    const int*   sp2      = (const int*)d_in[8];
    const int*   sp3      = (const int*)d_in[9];
    const float* conv3_w  = (const float*)d_in[10];
    const float* conv3_b  = (const float*)d_in[11];
    const float* conv2_w  = (const float*)d_in[12];
    const float* conv2_b  = (const float*)d_in[13];
    const float* conv1_w  = (const float*)d_in[14];
    const float* conv1_b  = (const float*)d_in[15];
    const float* conv0_w  = (const float*)d_in[16];
    const float* conv0_b  = (const float*)d_in[17];
    const float* convf_w  = (const float*)d_in[18];
    const float* convf_b  = (const float*)d_in[19];
    const int*   up0_cols = (const int*)d_in[21];
    const float* up0_vals = (const float*)d_in[22];
    const int*   up1_cols = (const int*)d_in[24];
    const float* up1_vals = (const float*)d_in[25];
    const int*   up2_cols = (const int*)d_in[27];
    const float* up2_vals = (const float*)d_in[28];
    const int*   up3_cols = (const int*)d_in[30];
    const float* up3_vals = (const float*)d_in[31];

    float* out      = (float*)d_out;
    float* out_main = out;                    // [8,20000,3] = 480000
    float* out_mu   = out + 480000;           // 512
    float* out_lv   = out + 480512;           // 512
    float* out_cls  = out + 481024;           // 8
    float* out_reg  = out + 481032;           // 8

    float* ws   = (float*)d_ws;
    float* bufB = ws;                         // max 8*20000*64 = 10,240,000 floats
    float* bufA = ws + 10240000;              // max 8*20000*32 =  5,120,000 floats
    float* zbuf = ws + 10240000 + 5120000;    // 1024 floats (padded 16x64)

    // 1. reparameterize + emit mu/logvar, zero-pad z rows 8..15
    z_kernel<<<4, 256, 0, stream>>>(idx, eps, mu_t, lv_t, zbuf, out_mu, out_lv);

    // 2. fc: [8,64] @ [64,20480] -> bufA ([8,80,256])
    fc_kernel<<<1280, 32, 0, stream>>>(zbuf, fc_w, fc_b, bufA);

    // 3. stage 3: pool (80->320) + conv 256->256  (M=2560 -> 80 blocks)
    pool_kernel<256><<<(NB * V3 * 256 + 255) / 256, 256, 0, stream>>>(
        bufA, up3_cols, up3_vals, bufB, V3, V4);
    spiral_conv<256, 256, V3, true><<<(NB * V3 + 31) / 32, 512, 0, stream>>>(
        bufB, sp3, conv3_w, conv3_b, bufA);

    // 4. stage 2: pool (320->1250) + conv 256->128  (M=10000 -> 313 blocks, tail)
    pool_kernel<256><<<(NB * V2 * 256 + 255) / 256, 256, 0, stream>>>(
        bufA, up2_cols, up2_vals, bufB, V2, V3);
    spiral_conv<256, 128, V2, true><<<(NB * V2 + 31) / 32, 256, 0, stream>>>(
        bufB, sp2, conv2_w, conv2_b, bufA);

    // 5. stage 1: pool (1250->5000) + conv 128->64  (M=40000 -> 1250 blocks)
    pool_kernel<128><<<(NB * V1 * 128 + 255) / 256, 256, 0, stream>>>(
        bufA, up1_cols, up1_vals, bufB, V1, V2);
    spiral_conv<128, 64, V1, true><<<(NB * V1 + 31) / 32, 128, 0, stream>>>(
        bufB, sp1, conv1_w, conv1_b, bufA);

    // 6. stage 0: pool (5000->20000) + conv 64->32  (M=160000 -> 5000 blocks)
    pool_kernel<64><<<(NB * V0 * 64 + 255) / 256, 256, 0, stream>>>(
        bufA, up0_cols, up0_vals, bufB, V0, V1);
    spiral_conv<64, 32, V0, true><<<(NB * V0 + 31) / 32, 64, 0, stream>>>(
        bufB, sp0, conv0_w, conv0_b, bufA);

    // 7. final conv 32->3 (no activation) -> d_out
    convf_kernel<<<(NB * V0 + 255) / 256, 256, 0, stream>>>(
        bufA, sp0, convf_w, convf_b, out_main);

    // 8. heads on z[:,0] (cls, sigmoid) and z[:,1] (reg)
    head_kernel<<<1, 32, 0, stream>>>(zbuf, 0,
        (const float*)d_in[32], (const float*)d_in[33], (const float*)d_in[34],
        (const float*)d_in[35], (const float*)d_in[36], (const float*)d_in[37],
        (const float*)d_in[38], (const float*)d_in[39], (const float*)d_in[40],
        (const float*)d_in[41], out_cls, 1);
    head_kernel<<<1, 32, 0, stream>>>(zbuf, 1,
        (const float*)d_in[42], (const float*)d_in[43], (const float*)d_in[44],
        (const float*)d_in[45], (const float*)d_in[46], (const float*)d_in[47],
        (const float*)d_in[48], (const float*)d_in[49], (const float*)d_in[50],
        (const float*)d_in[51], out_reg, 0);
}